// GINConv_68917045231840
// MI455X (gfx1250) — compile-verified
//
#include <hip/hip_runtime.h>

typedef float v2f __attribute__((ext_vector_type(2)));
typedef float v8f __attribute__((ext_vector_type(8)));

#define GIN_N   100000
#define GIN_E   1250000
#define GIN_D   64
#define WSTRIDE 68           // 64 + 4 pad: bank = (4*row + col) mod 64 -> conflict-free b64 frags

// ---------------------------------------------------------------- zero scratch
__global__ __launch_bounds__(256) void gin_zero(float4* __restrict__ ws, int n4) {
    int i = blockIdx.x * blockDim.x + threadIdx.x;
    if (i < n4) ws[i] = make_float4(0.f, 0.f, 0.f, 0.f);
}

// ---------------------------------------------------------------- edge scatter
// 16 threads per edge, float4 each: coalesced 256B gather of feat[src] and
// contiguous global_atomic_add_f32 into neigh[dst].
__global__ __launch_bounds__(256) void gin_scatter(const float* __restrict__ feat,
                                                   const int*   __restrict__ src,
                                                   const int*   __restrict__ dst,
                                                   float*       __restrict__ neigh) {
    int gid = blockIdx.x * blockDim.x + threadIdx.x;      // 20,000,000 = 78125*256 exactly
    int e = gid >> 4;
    int p = gid & 15;
    if (e >= GIN_E) return;
    int s = src[e];
    int d = dst[e];
    float4 v = *(const float4*)(feat + (long long)s * GIN_D + p * 4);
    float* o = neigh + (long long)d * GIN_D + p * 4;
    unsafeAtomicAdd(o + 0, v.x);
    unsafeAtomicAdd(o + 1, v.y);
    unsafeAtomicAdd(o + 2, v.z);
    unsafeAtomicAdd(o + 3, v.w);
}

// ---------------------------------------------------------------- fused combine + 2-layer MLP
// Block = 128 threads (4 waves). Each wave computes a 16x64 output tile with
// v_wmma_f32_16x16x4_f32: D = relu(((1+eps)*feat + neigh) @ W1 + b1) @ W2 + b2.
__global__ __launch_bounds__(128) void gin_mlp(const float* __restrict__ feat,
                                               const float* __restrict__ neigh,
                                               const float* __restrict__ W1,
                                               const float* __restrict__ b1,
                                               const float* __restrict__ W2,
                                               const float* __restrict__ b2,
                                               const float* __restrict__ epsP,
                                               float*       __restrict__ out,
                                               int numTiles) {
    __shared__ float sW1[GIN_D * WSTRIDE];        // W1^T, padded: sW1[n*WSTRIDE + k]
    __shared__ float sW2[GIN_D * WSTRIDE];        // W2^T, padded
    __shared__ float sA [4][16 * WSTRIDE];        // per-wave rst / h tile, padded

    const int tid = threadIdx.x;

    // cooperative transposed load of W1, W2 (coalesced global reads)
    for (int i = tid; i < GIN_D * GIN_D; i += 128) {
        int k = i >> 6;            // input dim (row of W)
        int n = i & 63;            // output dim (col of W)
        sW1[n * WSTRIDE + k] = W1[i];
        sW2[n * WSTRIDE + k] = W2[i];
    }
    __syncthreads();

    const int lane = tid & 31;
    const int wave = tid >> 5;
    const int tile = blockIdx.x * 4 + wave;       // 16-row tile index
    if (tile >= numTiles) return;                 // wave-uniform: EXEC stays all-ones

    const float es = 1.0f + epsP[0];
    float* A = &sA[wave][0];

    // ---- build rst tile (16 x 64) in LDS: each lane handles half a row
    {
        int r  = lane >> 1;
        int c0 = (lane & 1) * 32;
        long long g = (long long)(tile * 16 + r) * GIN_D + c0;
        #pragma unroll
        for (int j = 0; j < 8; ++j) {
            float4 f = *(const float4*)(feat  + g + j * 4);
            float4 n = *(const float4*)(neigh + g + j * 4);
            float4 v;
            v.x = fmaf(es, f.x, n.x);
            v.y = fmaf(es, f.y, n.y);
            v.z = fmaf(es, f.z, n.z);
            v.w = fmaf(es, f.w, n.w);
            *(float4*)&A[r * WSTRIDE + c0 + j * 4] = v;
        }
    }

    const int l15  = lane & 15;
    const int kb   = (lane >> 4) << 1;            // A/B frag K sub-offset: {0,2}
    const int arow = l15 * WSTRIDE;               // A frag: row M = lane&15
    const int mrow = (lane >> 4) << 3;            // C/D frag: M = v + 8*(lane>>4)

    // ---- GEMM 1: h = relu(rst @ W1 + b1)
    v8f acc[4];
    #pragma unroll
    for (int nt = 0; nt < 4; ++nt) {
        float bv = b1[nt * 16 + l15];             // bias depends only on N
        acc[nt] = (v8f){bv, bv, bv, bv, bv, bv, bv, bv};
    }
    #pragma unroll
    for (int k4 = 0; k4 < GIN_D; k4 += 4) {
        v2f a = *(const v2f*)&A[arow + k4 + kb];  // conflict-free ds_load_b64
        #pragma unroll
        for (int nt = 0; nt < 4; ++nt) {
            v2f b = *(const v2f*)&sW1[(nt * 16 + l15) * WSTRIDE + k4 + kb];
            acc[nt] = __builtin_amdgcn_wmma_f32_16x16x4_f32(
                false, a, false, b, (short)0, acc[nt], false, false);
        }
    }
    // ReLU + write h back into the wave-private LDS tile (same-wave DS ordering)
    #pragma unroll
    for (int nt = 0; nt < 4; ++nt)
        #pragma unroll
        for (int v = 0; v < 8; ++v)
            A[(mrow + v) * WSTRIDE + nt * 16 + l15] = fmaxf(acc[nt][v], 0.0f);

    // ---- GEMM 2: out = h @ W2 + b2
    #pragma unroll
    for (int nt = 0; nt < 4; ++nt) {
        float bv = b2[nt * 16 + l15];
        acc[nt] = (v8f){bv, bv, bv, bv, bv, bv, bv, bv};
    }
    #pragma unroll
    for (int k4 = 0; k4 < GIN_D; k4 += 4) {
        v2f a = *(const v2f*)&A[arow + k4 + kb];
        #pragma unroll
        for (int nt = 0; nt < 4; ++nt) {
            v2f b = *(const v2f*)&sW2[(nt * 16 + l15) * WSTRIDE + k4 + kb];
            acc[nt] = __builtin_amdgcn_wmma_f32_16x16x4_f32(
                false, a, false, b, (short)0, acc[nt], false, false);
        }
    }

    // ---- store 16x64 tile (16-lane groups write contiguous 64B)
    long long obase = (long long)tile * 16 * GIN_D;
    #pragma unroll
    for (int nt = 0; nt < 4; ++nt)
        #pragma unroll
        for (int v = 0; v < 8; ++v)
            out[obase + (long long)(mrow + v) * GIN_D + nt * 16 + l15] = acc[nt][v];
}

// ---------------------------------------------------------------- launcher
extern "C" void kernel_launch(void* const* d_in, const int* in_sizes, int n_in,
                              void* d_out, int out_size, void* d_ws, size_t ws_size,
                              hipStream_t stream) {
    const float* feat = (const float*)d_in[0];
    const float* W1   = (const float*)d_in[1];
    const float* b1   = (const float*)d_in[2];
    const float* W2   = (const float*)d_in[3];
    const float* b2   = (const float*)d_in[4];
    const float* eps  = (const float*)d_in[5];
    const int*   src  = (const int*)  d_in[6];
    const int*   dst  = (const int*)  d_in[7];
    float* out   = (float*)d_out;
    float* neigh = (float*)d_ws;                  // N*D*4 = 25.6 MB scratch

    // 1) zero neigh accumulator
    int n4 = GIN_N * GIN_D / 4;                   // 1,600,000 float4 = 6250 * 256
    gin_zero<<<n4 / 256, 256, 0, stream>>>((float4*)neigh, n4);

    // 2) edge scatter-add (memory/atomic bound: ~650MB traffic, the hot phase)
    int work = GIN_E * 16;                        // 20,000,000 = 78125 * 256
    gin_scatter<<<work / 256, 256, 0, stream>>>(feat, src, dst, neigh);

    // 3) fused combine + MLP via fp32 WMMA
    int numTiles = GIN_N / 16;                    // 6250
    int blocks   = (numTiles + 3) / 4;            // 1563
    gin_mlp<<<blocks, 128, 0, stream>>>(feat, neigh, W1, b1, W2, b2, eps, out, numTiles);
}